// GCNLayer_9405978378284
// MI455X (gfx1250) — compile-verified
//
#include <hip/hip_runtime.h>

typedef float v2f __attribute__((ext_vector_type(2)));
typedef float v8f __attribute__((ext_vector_type(8)));

#define DFEAT 128
#define LDS_STRIDE 132   // padded row stride (floats) to avoid LDS bank conflicts

// ---------------------------------------------------------------------------
// support[rowBase..rowBase+15][:] = X[rowBase..rowBase+15][:] @ W  (128x128)
// 256 threads = 8 waves; wave w computes the 16x16 output tile at cols 16w.
// Uses V_WMMA_F32_16X16X4_F32 (true fp32 WMMA) — GEMM is <1% of runtime, so
// we keep full fp32 precision instead of downcasting.
// ---------------------------------------------------------------------------
__global__ __launch_bounds__(256) void gcn_gemm(const float* __restrict__ X,
                                                const float* __restrict__ W,
                                                float* __restrict__ S,
                                                int N) {
  __shared__ float ldsX[16 * LDS_STRIDE];

  const int tid = threadIdx.x;
  const int rowBase = blockIdx.x << 4;
  if (rowBase >= N) return;

  // Cooperative, coalesced load of the 16x128 X tile into LDS (float4).
  const float4* X4 = (const float4*)(X + (size_t)rowBase * DFEAT);
#pragma unroll
  for (int j = 0; j < 2; ++j) {
    int q  = tid + j * 256;   // 0..511 float4 slots (16 rows x 32 float4)
    int r  = q >> 5;          // row in tile
    int c4 = q & 31;          // float4 column
    float4 v = X4[r * 32 + c4];
    *(float4*)&ldsX[r * LDS_STRIDE + c4 * 4] = v;
  }
  __syncthreads();

  const int lane = tid & 31;
  const int wave = tid >> 5;        // column tile 0..7
  const int colBase = wave << 4;
  const int m  = lane & 15;         // M index (A rows / C cols)
  const int kh = lane >> 4;         // K half select (ISA A layout: lanes 16-31 -> K+2)

  v8f acc = {};
  const float* arow = &ldsX[m * LDS_STRIDE];

#pragma unroll
  for (int k0 = 0; k0 < DFEAT; k0 += 4) {
    // A fragment 16x4: v0 = K=(k0 | k0+2), v1 = K=(k0+1 | k0+3) -> ds_load_b64
    v2f a = *(const v2f*)&arow[k0 + 2 * kh];
    // B fragment 4x16: rows striped across lanes within a VGPR
    v2f b;
    b.x = W[(size_t)(k0 + kh) * DFEAT + colBase + m];
    b.y = W[(size_t)(k0 + 2 + kh) * DFEAT + colBase + m];
    acc = __builtin_amdgcn_wmma_f32_16x16x4_f32(false, a, false, b,
                                                (short)0, acc, false, false);
  }

  // C/D layout: VGPR r -> row (r + 8*kh), col = m  (ISA 16x16 f32 layout)
  float* srow = S + (size_t)rowBase * DFEAT + colBase + m;
#pragma unroll
  for (int r = 0; r < 8; ++r) {
    srow[(size_t)(r + 8 * kh) * DFEAT] = acc[r];
  }
}

// ---------------------------------------------------------------------------
// out[src[e]][:] += val[e] * support[dst[e]][:]   (one wave32 per edge)
// support (25.6 MB/timestep) stays L2-resident -> gather + atomics hit L2.
// ---------------------------------------------------------------------------
__global__ __launch_bounds__(256) void gcn_scatter(const float* __restrict__ S,
                                                   const int* __restrict__ esrc,
                                                   const int* __restrict__ edst,
                                                   const float* __restrict__ eval,
                                                   float* __restrict__ out,
                                                   int E) {
  const int lane = threadIdx.x & 31;
  const int e = (blockIdx.x << 3) + (threadIdx.x >> 5);
  if (e >= E) return;

  // Edge indices are wave-uniform: force them scalar.
  const int s = __builtin_amdgcn_readfirstlane(esrc[e]);
  const int d = __builtin_amdgcn_readfirstlane(edst[e]);
  const float v = eval[e];

  const float4 msg = ((const float4*)(S + (size_t)d * DFEAT))[lane];
  float* orow = out + (size_t)s * DFEAT + lane * 4;
  atomicAdd(orow + 0, msg.x * v);
  atomicAdd(orow + 1, msg.y * v);
  atomicAdd(orow + 2, msg.z * v);
  atomicAdd(orow + 3, msg.w * v);
}

// ---------------------------------------------------------------------------
__global__ void zero_f4(float4* __restrict__ p, long n4) {
  long i = (long)blockIdx.x * blockDim.x + threadIdx.x;
  if (i < n4) p[i] = float4{0.f, 0.f, 0.f, 0.f};
}

__global__ void relu_f4(float4* __restrict__ p, long n4) {
  long i = (long)blockIdx.x * blockDim.x + threadIdx.x;
  if (i < n4) {
    float4 v = p[i];
    v.x = fmaxf(v.x, 0.f);
    v.y = fmaxf(v.y, 0.f);
    v.z = fmaxf(v.z, 0.f);
    v.w = fmaxf(v.w, 0.f);
    p[i] = v;
  }
}

// ---------------------------------------------------------------------------
extern "C" void kernel_launch(void* const* d_in, const int* in_sizes, int n_in,
                              void* d_out, int out_size, void* d_ws, size_t ws_size,
                              hipStream_t stream) {
  const float* X    = (const float*)d_in[0];   // [T, N, 128]
  const float* W    = (const float*)d_in[1];   // [128, 128]
  const int*   esrc = (const int*)  d_in[2];   // [T, E]
  const int*   edst = (const int*)  d_in[3];   // [T, E]
  const float* eval = (const float*)d_in[4];   // [T, E]
  float* out = (float*)d_out;                  // [T, N, 128]
  float* support = (float*)d_ws;               // [N, 128] scratch, reused per t

  const int T = 8;
  const int N = in_sizes[0] / (T * DFEAT);     // 50000
  const int E = in_sizes[2] / T;               // 800000

  // Zero the whole output (atomics accumulate into it; harness poisons).
  const long n4 = (long)T * N * DFEAT / 4;
  zero_f4<<<(unsigned)((n4 + 255) / 256), 256, 0, stream>>>((float4*)out, n4);

  const int gemmBlocks    = (N + 15) / 16;     // 3125 full 16-row tiles
  const int scatterBlocks = (E + 7) / 8;       // 8 waves (edges) per block

  for (int t = 0; t < T; ++t) {
    gcn_gemm<<<gemmBlocks, 256, 0, stream>>>(
        X + (size_t)t * N * DFEAT, W, support, N);
    gcn_scatter<<<scatterBlocks, 256, 0, stream>>>(
        support, esrc + (size_t)t * E, edst + (size_t)t * E,
        eval + (size_t)t * E, out + (size_t)t * N * DFEAT, E);
  }

  relu_f4<<<(unsigned)((n4 + 255) / 256), 256, 0, stream>>>((float4*)out, n4);
}